// ChannelAttentionModule_4088808866178
// MI455X (gfx1250) — compile-verified
//
#include <hip/hip_runtime.h>
#include <math.h>

typedef __attribute__((ext_vector_type(2))) float v2f;
typedef __attribute__((ext_vector_type(8))) float v8f;

// B=8, C=64, N=H*W=4096. One wave owns a 16-row query strip; a 128-thread
// block (4 waves) owns 64 rows and cooperatively stages 64x32 K/V slabs.
// Flash-attention streaming softmax; all GEMM work on V_WMMA_F32_16X16X4_F32.

// LDS floats: two slabs (V-major 64x34 + T-major 32x66) + 4 wave-private P bufs
#define SV_STRIDE 34          // [c][m] slab, even stride -> aligned v2f pairs
#define ST_STRIDE 66          // [m][c] slab, even stride -> aligned v2f pairs
#define SLAB_V    (64 * SV_STRIDE)          // 2176
#define SLAB_T    (32 * ST_STRIDE)          // 2112
#define SLAB      (SLAB_V + SLAB_T)         // 4288 per buffer
#define PB_STRIDE 34
#define PBUF      (16 * PB_STRIDE)          // 544 per wave
#define LDS_FLOATS (2 * SLAB + 4 * PBUF)    // 10752 floats = 43 KB

__global__ __launch_bounds__(128) void spatial_attn_f32_wmma(
    const float* __restrict__ x, float* __restrict__ out) {
  constexpr int C = 64;
  constexpr int N = 4096;

  const int tid  = threadIdx.x;
  const int wave = tid >> 5;
  const int lane = tid & 31;
  const int half = lane >> 4;
  const int j    = lane & 15;
  const int koff = 2 * half;            // K offset carried by upper lane half

  const int b      = blockIdx.x >> 6;
  const int rowblk = blockIdx.x & 63;
  const int n0     = rowblk * 64 + wave * 16;

  const float* __restrict__ xb = x + (size_t)b * C * N;

  __shared__ float lds[LDS_FLOATS];
  float* pbuf = lds + 2 * SLAB + wave * PBUF;   // wave-private P transpose buf

  // ---- Resident Q fragments (A-layout): A(M=j, K=c) = x[b, c, n0+j]
  v2f qf[16];
#pragma unroll
  for (int k4 = 0; k4 < 16; ++k4) {
    const int c0 = 4 * k4 + koff;
    qf[k4].x = xb[(size_t)c0 * N + n0 + j];
    qf[k4].y = xb[(size_t)(c0 + 1) * N + n0 + j];
  }

  // ---- Flash-attention state
  v8f zero = {};
  v8f o[4];
  o[0] = zero; o[1] = zero; o[2] = zero; o[3] = zero;
  float mrow[8], lrow[8];
#pragma unroll
  for (int r = 0; r < 8; ++r) { mrow[r] = -INFINITY; lrow[r] = 0.0f; }

  // Cooperative slab staging: x[b, 0:64, m0:m0+32] into both layouts.
  auto stage_slab = [&](int p, int m0) {
#pragma unroll 4
    for (int i = 0; i < 16; ++i) {
      const int e  = tid + 128 * i;     // 0..2047 over the 64x32 slab
      const int c  = e >> 5;
      const int mm = e & 31;
      const float v = xb[(size_t)c * N + m0 + mm];
      float* base = lds + p * SLAB;
      base[c * SV_STRIDE + mm] = v;                 // V-major [c][m]
      base[SLAB_V + mm * ST_STRIDE + c] = v;        // T-major [m][c]
    }
  };

  stage_slab(0, 0);
  __syncthreads();

  constexpr int ITERS = N / 32;         // 128 iterations of 32 key columns
  for (int it = 0; it < ITERS; ++it) {
    const int ping = it & 1;
    if (it + 1 < ITERS) stage_slab(ping ^ 1, (it + 1) * 32);

    const float* sV = lds + ping * SLAB;
    const float* sT = sV + SLAB_V;

    // ---- S = Q*K^T for two 16-col sub-tiles (32 chained f32 WMMAs)
    v8f a0 = zero, a1 = zero;
#pragma unroll
    for (int k4 = 0; k4 < 16; ++k4) {
      const int c0 = 4 * k4 + koff;
      const v2f bf0 = *(const v2f*)&sT[j * ST_STRIDE + c0];
      a0 = __builtin_amdgcn_wmma_f32_16x16x4_f32(
          false, qf[k4], false, bf0, (short)0, a0, false, false);
      const v2f bf1 = *(const v2f*)&sT[(16 + j) * ST_STRIDE + c0];
      a1 = __builtin_amdgcn_wmma_f32_16x16x4_f32(
          false, qf[k4], false, bf1, (short)0, a1, false, false);
    }

    // ---- Online softmax over both sub-tiles at once
    float p0[8], p1[8];
#pragma unroll
    for (int r = 0; r < 8; ++r) {
      float v = fmaxf(a0[r], a1[r]);
      v = fmaxf(v, __shfl_xor(v, 8, 32));
      v = fmaxf(v, __shfl_xor(v, 4, 32));
      v = fmaxf(v, __shfl_xor(v, 2, 32));
      v = fmaxf(v, __shfl_xor(v, 1, 32));
      const float mnew  = fmaxf(mrow[r], v);
      const float scale = __expf(mrow[r] - mnew);
      p0[r] = __expf(a0[r] - mnew);
      p1[r] = __expf(a1[r] - mnew);
      float s = p0[r] + p1[r];
      s += __shfl_xor(s, 8, 32);
      s += __shfl_xor(s, 4, 32);
      s += __shfl_xor(s, 2, 32);
      s += __shfl_xor(s, 1, 32);
      lrow[r] = lrow[r] * scale + s;
      mrow[r] = mnew;
      o[0][r] *= scale; o[1][r] *= scale; o[2][r] *= scale; o[3][r] *= scale;
    }

    // ---- P (16x32): D-layout -> wave-private LDS -> A-layout fragments
#pragma unroll
    for (int r = 0; r < 8; ++r) {
      pbuf[(r + 8 * half) * PB_STRIDE + j]      = p0[r];
      pbuf[(r + 8 * half) * PB_STRIDE + 16 + j] = p1[r];
    }
    v2f pf[8];
#pragma unroll
    for (int k4 = 0; k4 < 8; ++k4)
      pf[k4] = *(const v2f*)&pbuf[j * PB_STRIDE + 4 * k4 + koff];

    // ---- O += P*V over K=32 (32 chained f32 WMMAs); V(K=m, N=c)=sV[c][m]
#pragma unroll
    for (int cg = 0; cg < 4; ++cg) {
      v8f oc = o[cg];
      const float* vb = &sV[(16 * cg + j) * SV_STRIDE + koff];
#pragma unroll
      for (int k4 = 0; k4 < 8; ++k4) {
        const v2f vf = *(const v2f*)&vb[4 * k4];
        oc = __builtin_amdgcn_wmma_f32_16x16x4_f32(
            false, pf[k4], false, vf, (short)0, oc, false, false);
      }
      o[cg] = oc;
    }

    __syncthreads();   // protects shared slab reuse across waves
  }

  // ---- Normalize, transpose through wave-private LDS, coalesced store
#pragma unroll
  for (int r = 0; r < 8; ++r) {
    const float inv = 1.0f / lrow[r];
    o[0][r] *= inv; o[1][r] *= inv; o[2][r] *= inv; o[3][r] *= inv;
  }
  float* eb = lds + wave * (64 * 17);   // safe after final barrier
#pragma unroll
  for (int cg = 0; cg < 4; ++cg)
#pragma unroll
    for (int r = 0; r < 8; ++r)
      eb[(16 * cg + j) * 17 + (r + 8 * half)] = o[cg][r];

  float* __restrict__ outb = out + (size_t)b * C * N;
#pragma unroll 4
  for (int i = 0; i < 32; ++i) {
    const int c = 2 * i + half;
    const size_t gi = (size_t)c * N + n0 + j;
    outb[gi] = xb[gi] + eb[c * 17 + j];
  }
}

extern "C" void kernel_launch(void* const* d_in, const int* in_sizes, int n_in,
                              void* d_out, int out_size, void* d_ws, size_t ws_size,
                              hipStream_t stream) {
  (void)in_sizes; (void)n_in; (void)d_ws; (void)ws_size; (void)out_size;
  const float* x = (const float*)d_in[0];
  float* out = (float*)d_out;
  dim3 grid(512), block(128);
  spatial_attn_f32_wmma<<<grid, block, 0, stream>>>(x, out);
}